// RobustSAE_35622458753285
// MI455X (gfx1250) — compile-verified
//
#include <hip/hip_runtime.h>

typedef float v2f __attribute__((ext_vector_type(2)));
typedef float v8f __attribute__((ext_vector_type(8)));

#define D_MODEL 768
#define D_SAE   32768
#define BATCH   4096
#define TOPK    32
#define NSTAGE  (D_MODEL / 16)   // 48 K-stages of 16

// LDS strides chosen for 16B alignment (async B128 writes) + bank-conflict-free reads
#define A_STRIDE 20   // 80 B/row: r*20 mod 64 distinct for r=0..15; halves differ mod 4
#define B_STRIDE 72   // 288 B/row: k-halves (rows +2) occupy disjoint 16-bank windows

// ---------------------------------------------------------------------------
// Kernel 1: z_pre = x @ W_enc + b_enc  via V_WMMA_F32_16X16X4_F32 (exact f32)
// Block: 256 threads (8 waves), C tile 128x128; each wave owns 64x32 (eight
// 16x16 accumulators). Staging: GLOBAL_LOAD_ASYNC_TO_LDS_B128, DOUBLE-BUFFERED
// on ASYNCcnt: issue stage s+1, then s_wait_asynccnt 4 (in-order completion)
// so stage s is ready while the prefetch stays in flight.
// ---------------------------------------------------------------------------
__global__ __launch_bounds__(256)
void sae_encode_wmma(const float* __restrict__ x,
                     const float* __restrict__ W_enc,
                     const float* __restrict__ b_enc,
                     float* __restrict__ z_pre) {
    __shared__ float A_s[2][128 * A_STRIDE];   // 2 x (128 rows x 16 K)
    __shared__ float B_s[2][16 * B_STRIDE];    // 2 x (16 K x 128 cols)

    const int tid    = threadIdx.x;
    const int lane   = tid & 31;
    const int wave   = tid >> 5;
    const int wm     = wave >> 2;    // 0..1  (M region of 64 rows)
    const int wn     = wave & 3;     // 0..3  (N region of 32 cols)
    const int nlocal = lane & 15;
    const int khalf  = lane >> 4;

    const int mBase = blockIdx.y * 128;
    const int nBase = blockIdx.x * 128;

    v8f acc[4][2];
    #pragma unroll
    for (int mt = 0; mt < 4; mt++)
        #pragma unroll
        for (int nt = 0; nt < 2; nt++) acc[mt][nt] = (v8f){};

    // wave-relative LDS byte offsets (low 32 bits of generic pointer)
    const unsigned ldsA0 = (unsigned)(unsigned long long)(void*)&A_s[0][0];
    const unsigned ldsB0 = (unsigned)(unsigned long long)(void*)&B_s[0][0];
    const unsigned aBufBytes = 128 * A_STRIDE * 4;
    const unsigned bBufBytes = 16 * B_STRIDE * 4;

    // issue one stage's async staging: 4 B128 per thread (A:2, B:2)
    auto issue_stage = [&](int k0, int buf) {
        #pragma unroll
        for (int p = 0; p < 2; p++) {
            const int c  = tid + p * 256;
            const int ar = c >> 2, ac = (c & 3) * 4;
            const unsigned lA = ldsA0 + buf * aBufBytes
                              + (unsigned)((ar * A_STRIDE + ac) * 4);
            const unsigned long long gA =
                (unsigned long long)&x[(size_t)(mBase + ar) * D_MODEL + k0 + ac];
            asm volatile("global_load_async_to_lds_b128 %0, %1, off"
                         :: "v"(lA), "v"(gA) : "memory");
            const int br = c >> 5, bc = (c & 31) * 4;
            const unsigned lB = ldsB0 + buf * bBufBytes
                              + (unsigned)((br * B_STRIDE + bc) * 4);
            const unsigned long long gB =
                (unsigned long long)&W_enc[(size_t)(k0 + br) * D_SAE + nBase + bc];
            asm volatile("global_load_async_to_lds_b128 %0, %1, off"
                         :: "v"(lB), "v"(gB) : "memory");
        }
    };

    issue_stage(0, 0);   // prologue

    for (int s = 0; s < NSTAGE; s++) {
        const int buf = s & 1;
        if (s + 1 < NSTAGE) {
            issue_stage((s + 1) * 16, buf ^ 1);
            asm volatile("s_wait_asynccnt 0x4" ::: "memory");  // stage s landed
        } else {
            asm volatile("s_wait_asynccnt 0x0" ::: "memory");
        }
        __syncthreads();

        const float* Ab = &A_s[buf][0];
        const float* Bb = &B_s[buf][0];
        #pragma unroll
        for (int kk = 0; kk < 4; kk++) {
            const int kb = kk * 4 + khalf * 2;
            v2f b0, b1;
            b0.x = Bb[(kb + 0) * B_STRIDE + wn * 32 + nlocal];
            b0.y = Bb[(kb + 1) * B_STRIDE + wn * 32 + nlocal];
            b1.x = Bb[(kb + 0) * B_STRIDE + wn * 32 + 16 + nlocal];
            b1.y = Bb[(kb + 1) * B_STRIDE + wn * 32 + 16 + nlocal];
            #pragma unroll
            for (int mt = 0; mt < 4; mt++) {
                const int ra = (wm * 64 + mt * 16 + nlocal) * A_STRIDE + kb;
                v2f a;
                a.x = Ab[ra + 0];
                a.y = Ab[ra + 1];
                acc[mt][0] = __builtin_amdgcn_wmma_f32_16x16x4_f32(
                    false, a, false, b0, (short)0, acc[mt][0], false, false);
                acc[mt][1] = __builtin_amdgcn_wmma_f32_16x16x4_f32(
                    false, a, false, b1, (short)0, acc[mt][1], false, false);
            }
        }
        __syncthreads();
    }

    // epilogue: C/D layout VGPR i -> row (i + khalf*8), col = nlocal
    #pragma unroll
    for (int nt = 0; nt < 2; nt++) {
        const int gcol = nBase + wn * 32 + nt * 16 + nlocal;
        const float bias = b_enc[gcol];
        #pragma unroll
        for (int mt = 0; mt < 4; mt++) {
            #pragma unroll
            for (int i = 0; i < 8; i++) {
                const int r = mBase + wm * 64 + mt * 16 + khalf * 8 + i;
                z_pre[(size_t)r * D_SAE + gcol] = acc[mt][nt][i] + bias;
            }
        }
    }
}

// ---------------------------------------------------------------------------
// Kernel 2: per-row (4096 blocks, 256 threads):
//   1) async-stage raw z_pre row into LDS (128 KB < 320 KB WGP LDS)
//   2) peeled pass 0: ReLU in place + top-8-bit histogram
//   3) passes 1..3: MSD radix refine -> exact k-th largest bit pattern
//   4) write dense z_sparse row (threshold + capped tie handling)
//   5) fused sparse decode: x_recon = sum_k val*W_dec[idx,:] + b_dec
// ---------------------------------------------------------------------------
__global__ __launch_bounds__(256)
void sae_topk_decode(const float* __restrict__ z_pre,
                     const float* __restrict__ W_dec,
                     const float* __restrict__ b_dec,
                     float* __restrict__ z_sparse,
                     float* __restrict__ x_recon) {
    extern __shared__ unsigned char smem_raw[];
    float*    zrow    = (float*)smem_raw;              // D_SAE floats
    unsigned* hist    = (unsigned*)(zrow + D_SAE);     // 256 bins
    float*    sel_val = (float*)(hist + 256);          // TOPK
    unsigned* sel_idx = (unsigned*)(sel_val + TOPK);   // TOPK
    unsigned* s_scal  = (unsigned*)(sel_idx + TOPK);   // [0]=prefix [1]=cntAbove [2]=selCnt [3]=eqTaken

    const int tid = threadIdx.x;
    const int row = blockIdx.x;
    const float* zp = z_pre    + (size_t)row * D_SAE;
    float*       zs = z_sparse + (size_t)row * D_SAE;

    // async-stage the raw row straight into LDS (32 B128 chunks per thread)
    {
        const unsigned ldsZ = (unsigned)(unsigned long long)(void*)zrow;
        #pragma unroll 4
        for (int i = 0; i < D_SAE / 4 / 256; i++) {
            const int c = tid + i * 256;
            const unsigned l = ldsZ + (unsigned)c * 16u;
            const unsigned long long g = (unsigned long long)(zp + (size_t)c * 4);
            asm volatile("global_load_async_to_lds_b128 %0, %1, off"
                         :: "v"(l), "v"(g) : "memory");
        }
    }
    // init control state while the DMA is in flight (disjoint LDS region)
    hist[tid] = 0u;
    if (tid < 4) s_scal[tid] = 0u;
    if (tid < TOPK) { sel_val[tid] = 0.f; sel_idx[tid] = 0u; }
    asm volatile("s_wait_asynccnt 0x0" ::: "memory");
    __syncthreads();

    // pass 0 (peeled): ReLU in place + histogram of top 8 bits
    for (int i = 0; i < D_SAE / 256; i++) {
        const int idx = tid + i * 256;
        float v = zrow[idx];
        v = v > 0.f ? v : 0.f;
        zrow[idx] = v;
        atomicAdd(&hist[__float_as_uint(v) >> 24], 1u);
    }
    __syncthreads();
    if (tid == 0) {
        unsigned cum = 0; int b = 255;
        for (; b >= 0; b--) {
            const unsigned h = hist[b];
            if (cum + h >= TOPK) break;
            cum += h;
        }
        s_scal[0] = (unsigned)(b < 0 ? 0 : b);
        s_scal[1] = cum;
    }
    __syncthreads();

    // passes 1..3: refine 8 bits at a time
    for (int pass = 1; pass < 4; pass++) {
        hist[tid] = 0u;
        __syncthreads();
        const int shift = 24 - 8 * pass;
        const unsigned pref = s_scal[0];
        for (int i = 0; i < D_SAE / 256; i++) {
            const unsigned b = __float_as_uint(zrow[tid + i * 256]);
            if ((b >> (shift + 8)) == pref)
                atomicAdd(&hist[(b >> shift) & 0xFFu], 1u);
        }
        __syncthreads();
        if (tid == 0) {
            const unsigned need = TOPK - s_scal[1];
            unsigned cum = 0; int b = 255;
            for (; b >= 0; b--) {
                const unsigned h = hist[b];
                if (cum + h >= need) break;
                cum += h;
            }
            s_scal[0] = (pref << 8) | (unsigned)(b < 0 ? 0 : b);
            s_scal[1] += cum;
        }
        __syncthreads();
    }

    const unsigned T      = s_scal[0];          // k-th largest bit pattern
    const unsigned needEq = TOPK - s_scal[1];   // ties at T to admit

    // write dense sparse code + collect selected (val, idx)
    for (int i = 0; i < D_SAE / 256; i++) {
        const int idx = tid + i * 256;
        const float v = zrow[idx];
        const unsigned b = __float_as_uint(v);
        float outv = 0.f;
        if (b > T) {
            outv = v;
            const unsigned s = atomicAdd(&s_scal[2], 1u);
            if (s < TOPK) { sel_val[s] = v; sel_idx[s] = (unsigned)idx; }
        } else if (b == T) {
            const unsigned t = atomicAdd(&s_scal[3], 1u);
            if (t < needEq) {
                outv = v;
                const unsigned s = atomicAdd(&s_scal[2], 1u);
                if (s < TOPK) { sel_val[s] = v; sel_idx[s] = (unsigned)idx; }
            }
        }
        zs[idx] = outv;
    }
    __syncthreads();

    // fused sparse decode: each thread owns 3 of 768 output columns
    float a0 = 0.f, a1 = 0.f, a2 = 0.f;
    const int c0 = tid, c1 = tid + 256, c2 = tid + 512;
    #pragma unroll 8
    for (int j = 0; j < TOPK; j++) {
        const float val = sel_val[j];
        const float* wd = W_dec + (size_t)sel_idx[j] * D_MODEL;
        a0 += val * wd[c0];
        a1 += val * wd[c1];
        a2 += val * wd[c2];
    }
    float* xr = x_recon + (size_t)row * D_MODEL;
    xr[c0] = a0 + b_dec[c0];
    xr[c1] = a1 + b_dec[c1];
    xr[c2] = a2 + b_dec[c2];
}

// ---------------------------------------------------------------------------
extern "C" void kernel_launch(void* const* d_in, const int* in_sizes, int n_in,
                              void* d_out, int out_size, void* d_ws, size_t ws_size,
                              hipStream_t stream) {
    (void)in_sizes; (void)n_in; (void)out_size; (void)d_ws; (void)ws_size;
    const float* x     = (const float*)d_in[0];
    const float* W_enc = (const float*)d_in[1];
    const float* b_enc = (const float*)d_in[2];
    const float* W_dec = (const float*)d_in[3];
    const float* b_dec = (const float*)d_in[4];
    // d_in[5] = topk (fixed at 32)

    float* out      = (float*)d_out;
    float* x_recon  = out;                                   // [B, 768]
    float* z_sparse = out + (size_t)BATCH * D_MODEL;         // [B, 32768]
    float* z_pre    = z_sparse + (size_t)BATCH * D_SAE;      // [B, 32768]

    dim3 g1(D_SAE / 128, BATCH / 128);
    sae_encode_wmma<<<g1, 256, 0, stream>>>(x, W_enc, b_enc, z_pre);

    const size_t shmem = (size_t)D_SAE * 4 + 256 * 4 + TOPK * 4 + TOPK * 4 + 16;
    (void)hipFuncSetAttribute(reinterpret_cast<const void*>(sae_topk_decode),
                              hipFuncAttributeMaxDynamicSharedMemorySize, (int)shmem);
    sae_topk_decode<<<BATCH, 256, shmem, stream>>>(z_pre, W_dec, b_dec, z_sparse, x_recon);
}